// PointNetSetAbstraction_32409823215992
// MI455X (gfx1250) — compile-verified
//
#include <hip/hip_runtime.h>
#include <hip/hip_bf16.h>

typedef float v2f __attribute__((ext_vector_type(2)));
typedef float v8f __attribute__((ext_vector_type(8)));

#define B_   16
#define N_   4096
#define M_   1024
#define S_   32
#define NJ_  (B_ * M_ * S_)      /* 524288 columns of the per-layer GEMM */
#define RADIUS2_ 0.04f
#define BN_EPS_  1e-5f

// ---------------------------------------------------------------------------
// Farthest point sampling: one 1024-thread block per batch. Coords + running
// min-distance live in registers (4 pts/thread); argmax via wave shuffle
// reduction + 32-entry LDS reduction. Records `far` BEFORE update (matches
// the scan semantics: idx[0] == 0). Also writes new_xyz into d_out.
// ---------------------------------------------------------------------------
__global__ __launch_bounds__(1024)
void fps_kernel(const float* __restrict__ xyz, int* __restrict__ fps_idx,
                float* __restrict__ out_newxyz)
{
  const int b    = blockIdx.x;
  const int tid  = threadIdx.x;
  const int lane = tid & 31;
  const int wave = tid >> 5;
  const float* px = xyz + (size_t)b * 3 * N_;
  const float* py = px + N_;
  const float* pz = py + N_;

  const int T = 1024;
  const int PPT = N_ / T; // 4 points per thread
  float lx[PPT], ly[PPT], lz[PPT], dist[PPT];
#pragma unroll
  for (int k = 0; k < PPT; ++k) {
    int n = k * T + tid;
    lx[k] = px[n]; ly[k] = py[n]; lz[k] = pz[n];
    dist[k] = 1e10f;
  }

  __shared__ float s_val[32];
  __shared__ int   s_idx[32];
  __shared__ float s_c[3];
  __shared__ int   s_far;

  int far = 0;
  for (int it = 0; it < M_; ++it) {
    if (tid == 0) {
      s_c[0] = px[far]; s_c[1] = py[far]; s_c[2] = pz[far];
      fps_idx[b * M_ + it] = far;
    }
    __syncthreads();
    const float cx = s_c[0], cy = s_c[1], cz = s_c[2];
    float bv = -1.0f; int bi = 0;
#pragma unroll
    for (int k = 0; k < PPT; ++k) {
      float dx = lx[k] - cx, dy = ly[k] - cy, dz = lz[k] - cz;
      float d  = dx * dx + dy * dy + dz * dz;
      float nd = fminf(dist[k], d);
      dist[k] = nd;
      int n = k * T + tid;
      if (nd > bv) { bv = nd; bi = n; }   // strict > keeps lowest n in-thread
    }
#pragma unroll
    for (int off = 16; off >= 1; off >>= 1) {
      float ov = __shfl_xor(bv, off, 32);
      int   oi = __shfl_xor(bi, off, 32);
      if (ov > bv || (ov == bv && oi < bi)) { bv = ov; bi = oi; }
    }
    if (lane == 0) { s_val[wave] = bv; s_idx[wave] = bi; }
    __syncthreads();
    if (wave == 0) {
      bv = s_val[lane]; bi = s_idx[lane];
#pragma unroll
      for (int off = 16; off >= 1; off >>= 1) {
        float ov = __shfl_xor(bv, off, 32);
        int   oi = __shfl_xor(bi, off, 32);
        if (ov > bv || (ov == bv && oi < bi)) { bv = ov; bi = oi; }
      }
      if (lane == 0) s_far = bi;
    }
    __syncthreads();
    far = s_far;
  }
  __syncthreads();
  for (int m = tid; m < M_; m += T) {
    int i = fps_idx[b * M_ + m];
    out_newxyz[((size_t)b * 3 + 0) * M_ + m] = px[i];
    out_newxyz[((size_t)b * 3 + 1) * M_ + m] = py[i];
    out_newxyz[((size_t)b * 3 + 2) * M_ + m] = pz[i];
  }
}

// ---------------------------------------------------------------------------
// Ball query + grouping: one wave per center. Scan the 4096 points in 32-wide
// chunks in index order; __ballot + prefix popcount assigns in-radius points
// to slots 0..31 exactly like sort(where(d2<=r2, idx, N))[:32]. Pads short
// groups with slot 0. Directly builds the GEMM input X0 (8 channels: 3 norm
// coords, 3 point feats, 2 zero-pad channels so Cin is a multiple of 4).
// ---------------------------------------------------------------------------
__global__ __launch_bounds__(256)
void ballquery_group_kernel(const float* __restrict__ xyz,
                            const float* __restrict__ points,
                            const float* __restrict__ newxyz,
                            float* __restrict__ x0)
{
  const int lane   = threadIdx.x & 31;
  const int wave   = threadIdx.x >> 5;
  const int center = blockIdx.x * 8 + wave;       // 0 .. B*M-1
  const int b = center >> 10;
  const int m = center & (M_ - 1);
  const float* px = xyz    + (size_t)b * 3 * N_;
  const float* py = px + N_;
  const float* pz = py + N_;
  const float* qx = points + (size_t)b * 3 * N_;
  const float* qy = qx + N_;
  const float* qz = qy + N_;
  const float cx = newxyz[((size_t)b * 3 + 0) * M_ + m];
  const float cy = newxyz[((size_t)b * 3 + 1) * M_ + m];
  const float cz = newxyz[((size_t)b * 3 + 2) * M_ + m];

  __shared__ int s_sel[8][S_];
  int count = 0;  // wave-uniform
  for (int chunk = 0; chunk < N_ / 32 && count < S_; ++chunk) {
    int n = chunk * 32 + lane;
    float dx = px[n] - cx, dy = py[n] - cy, dz = pz[n] - cz;
    bool in = (dx * dx + dy * dy + dz * dz) <= RADIUS2_;
    unsigned mask = (unsigned)__ballot(in);   // wave32: low 32 bits
    if (in) {
      int slot = count + __popc(mask & ((1u << lane) - 1u));
      if (slot < S_) s_sel[wave][slot] = n;
    }
    count += __popc(mask);
  }
  __syncthreads();
  const int cnt = count < S_ ? count : S_;     // center itself => cnt >= 1
  const int i = (lane < cnt) ? s_sel[wave][lane] : s_sel[wave][0];
  const size_t j = (size_t)center * S_ + lane;
  x0[(size_t)0 * NJ_ + j] = px[i] - cx;
  x0[(size_t)1 * NJ_ + j] = py[i] - cy;
  x0[(size_t)2 * NJ_ + j] = pz[i] - cz;
  x0[(size_t)3 * NJ_ + j] = qx[i];
  x0[(size_t)4 * NJ_ + j] = qy[i];
  x0[(size_t)5 * NJ_ + j] = qz[i];
  x0[(size_t)6 * NJ_ + j] = 0.0f;
  x0[(size_t)7 * NJ_ + j] = 0.0f;
}

// ---------------------------------------------------------------------------
// Zero-pad W (Cout x Cin) -> Wp (Cout x CinP) so K can step by 4.
// ---------------------------------------------------------------------------
__global__ void pad_w_kernel(const float* __restrict__ W, float* __restrict__ Wp,
                             int Cout, int Cin, int CinP)
{
  int i = blockIdx.x * blockDim.x + threadIdx.x;
  if (i < Cout * CinP) {
    int o = i / CinP, c = i % CinP;
    Wp[i] = (c < Cin) ? W[o * Cin + c] : 0.0f;
  }
}

// ---------------------------------------------------------------------------
// Y[Cout x NJ] = W[Cout x Cin] * X[Cin x NJ] + bias, via V_WMMA_F32_16X16X4_F32.
// VGPR layouts per CDNA5 ISA 7.12.2:
//   A (16x4 f32): lane l<16 row M=l holds K=c0,c0+1 in a.x/a.y; lanes 16-31
//                 hold K=c0+2,c0+3 for the same rows.
//   B (4x16 f32): vgpr v, half h -> row K = v + 2h, col N = lane&15.
//   C/D (16x16):  vgpr r, half h -> row o0 + r + 8h, col N = lane&15.
// Each wave owns a 16(out-ch) x 64(col) tile: one A fragment feeds 4 WMMAs
// per K-step (4x A reuse), 8 waves/block, EXEC all-ones at every WMMA.
// ---------------------------------------------------------------------------
__global__ __launch_bounds__(256)
void mlp_matmul_wmma(const float* __restrict__ X, const float* __restrict__ W,
                     const float* __restrict__ bias, float* __restrict__ Y,
                     int Cin)
{
  const int lane = threadIdx.x & 31;
  const int wave = threadIdx.x >> 5;
  const int half = lane >> 4;
  const int l    = lane & 15;
  const int j0   = (blockIdx.x * 8 + wave) * 64;   // 4 x 16-col tiles per wave
  const int o0   = blockIdx.y * 16;

  v8f acc0, acc1, acc2, acc3;
#pragma unroll
  for (int r = 0; r < 8; ++r) {
    float bv = bias[o0 + r + half * 8];
    acc0[r] = bv; acc1[r] = bv; acc2[r] = bv; acc3[r] = bv;
  }

  const float* wrow = W + (size_t)(o0 + l) * Cin;
  for (int c0 = 0; c0 < Cin; c0 += 4) {
    v2f a;
    a.x = wrow[c0 + half * 2 + 0];
    a.y = wrow[c0 + half * 2 + 1];
    const size_t r0 = (size_t)(c0 + half * 2 + 0) * NJ_ + j0 + l;
    const size_t r1 = (size_t)(c0 + half * 2 + 1) * NJ_ + j0 + l;
    v2f b0, b1, b2, b3;
    b0.x = X[r0 +  0]; b0.y = X[r1 +  0];
    b1.x = X[r0 + 16]; b1.y = X[r1 + 16];
    b2.x = X[r0 + 32]; b2.y = X[r1 + 32];
    b3.x = X[r0 + 48]; b3.y = X[r1 + 48];
    acc0 = __builtin_amdgcn_wmma_f32_16x16x4_f32(
        false, a, false, b0, (short)0, acc0, false, false);
    acc1 = __builtin_amdgcn_wmma_f32_16x16x4_f32(
        false, a, false, b1, (short)0, acc1, false, false);
    acc2 = __builtin_amdgcn_wmma_f32_16x16x4_f32(
        false, a, false, b2, (short)0, acc2, false, false);
    acc3 = __builtin_amdgcn_wmma_f32_16x16x4_f32(
        false, a, false, b3, (short)0, acc3, false, false);
  }
#pragma unroll
  for (int r = 0; r < 8; ++r) {
    const size_t row = (size_t)(o0 + r + half * 8) * NJ_ + j0 + l;
    Y[row +  0] = acc0[r];
    Y[row + 16] = acc1[r];
    Y[row + 32] = acc2[r];
    Y[row + 48] = acc3[r];
  }
}

// ---------------------------------------------------------------------------
// Per-channel sum / sum-of-squares over all NJ columns (BN over axes 0,2,3).
// grid = (Cout, SEGS); block-local shuffle+LDS reduce, then 2 atomics/block.
// ---------------------------------------------------------------------------
__global__ __launch_bounds__(256)
void channel_stats_kernel(const float* __restrict__ Y, float* __restrict__ sums,
                          float* __restrict__ sumsq)
{
  const int c   = blockIdx.x;
  const int per = NJ_ / gridDim.y;
  const float* p = Y + (size_t)c * NJ_ + (size_t)blockIdx.y * per;
  float s = 0.0f, s2 = 0.0f;
  for (int i = threadIdx.x; i < per; i += blockDim.x) {
    float v = p[i];
    s += v; s2 += v * v;
  }
#pragma unroll
  for (int off = 16; off >= 1; off >>= 1) {
    s  += __shfl_xor(s, off, 32);
    s2 += __shfl_xor(s2, off, 32);
  }
  __shared__ float ss[8], ss2[8];
  const int lane = threadIdx.x & 31, wave = threadIdx.x >> 5;
  if (lane == 0) { ss[wave] = s; ss2[wave] = s2; }
  __syncthreads();
  if (threadIdx.x == 0) {
    float t = 0.0f, t2 = 0.0f;
    for (int k = 0; k < 8; ++k) { t += ss[k]; t2 += ss2[k]; }
    atomicAdd(&sums[c], t);
    atomicAdd(&sumsq[c], t2);
  }
}

__global__ void zero_kernel(float* __restrict__ p, int n)
{
  int i = blockIdx.x * blockDim.x + threadIdx.x;
  if (i < n) p[i] = 0.0f;
}

// In-place BN + ReLU (layers 1 and 2). grid = (512, Cout).
__global__ __launch_bounds__(256)
void bn_relu_kernel(float* __restrict__ Y, const float* __restrict__ sums,
                    const float* __restrict__ sumsq,
                    const float* __restrict__ gamma, const float* __restrict__ beta)
{
  const int c = blockIdx.y;
  const float inv   = 1.0f / (float)NJ_;
  const float mean  = sums[c] * inv;
  const float var   = sumsq[c] * inv - mean * mean;
  const float scale = gamma[c] * rsqrtf(var + BN_EPS_);
  const float shift = beta[c] - mean * scale;
  float* p = Y + (size_t)c * NJ_;
  for (int i = blockIdx.x * blockDim.x + threadIdx.x; i < NJ_;
       i += gridDim.x * blockDim.x) {
    float v = p[i] * scale + shift;
    p[i] = v > 0.0f ? v : 0.0f;
  }
}

// Layer-3 epilogue: BN + ReLU + max over the S=32 group, straight to d_out.
// grid = (B*M/256, 128); each thread streams 32 contiguous floats.
__global__ __launch_bounds__(256)
void bn_relu_max_kernel(const float* __restrict__ Y, const float* __restrict__ sums,
                        const float* __restrict__ sumsq,
                        const float* __restrict__ gamma, const float* __restrict__ beta,
                        float* __restrict__ out)
{
  const int o  = blockIdx.y;
  const int bm = blockIdx.x * blockDim.x + threadIdx.x;
  const float inv   = 1.0f / (float)NJ_;
  const float mean  = sums[o] * inv;
  const float var   = sumsq[o] * inv - mean * mean;
  const float scale = gamma[o] * rsqrtf(var + BN_EPS_);
  const float shift = beta[o] - mean * scale;
  const float* p = Y + (size_t)o * NJ_ + (size_t)bm * S_;
  float mx = 0.0f;  // ReLU outputs are >= 0
#pragma unroll 4
  for (int s = 0; s < S_; ++s) {
    float v = p[s] * scale + shift;
    v = v > 0.0f ? v : 0.0f;
    if (v > mx) mx = v;
  }
  const int b = bm >> 10, m = bm & (M_ - 1);
  out[(size_t)B_ * 3 * M_ + (((size_t)b * 128 + o) << 10) + m] = mx;
}

// ---------------------------------------------------------------------------
extern "C" void kernel_launch(void* const* d_in, const int* in_sizes, int n_in,
                              void* d_out, int out_size, void* d_ws, size_t ws_size,
                              hipStream_t stream)
{
  const float* xyz    = (const float*)d_in[0];
  const float* points = (const float*)d_in[1];
  const float* W1  = (const float*)d_in[2];
  const float* b1  = (const float*)d_in[3];
  const float* g1  = (const float*)d_in[4];
  const float* be1 = (const float*)d_in[5];
  const float* W2  = (const float*)d_in[6];
  const float* b2  = (const float*)d_in[7];
  const float* g2  = (const float*)d_in[8];
  const float* be2 = (const float*)d_in[9];
  const float* W3  = (const float*)d_in[10];
  const float* b3  = (const float*)d_in[11];
  const float* g3  = (const float*)d_in[12];
  const float* be3 = (const float*)d_in[13];
  float* out = (float*)d_out;

  // Workspace carve-up (~554 MB total, 256B-aligned regions).
  char* w = (char*)d_ws;
  auto take = [&](size_t bytes) -> char* {
    char* p = w; w += (bytes + 255) & ~(size_t)255; return p;
  };
  int*   fps_idx = (int*)  take((size_t)B_ * M_ * sizeof(int));
  float* stats   = (float*)take(256 * sizeof(float));       // sums[128]+sumsq[128]
  float* wpad    = (float*)take(64 * 8 * sizeof(float));
  float* x0      = (float*)take((size_t)8   * NJ_ * sizeof(float));
  float* yA      = (float*)take((size_t)64  * NJ_ * sizeof(float));
  float* yB      = (float*)take((size_t)64  * NJ_ * sizeof(float));
  float* yC      = (float*)take((size_t)128 * NJ_ * sizeof(float));
  float* sums  = stats;
  float* sumsq = stats + 128;

  fps_kernel<<<B_, 1024, 0, stream>>>(xyz, fps_idx, out);
  ballquery_group_kernel<<<(B_ * M_) / 8, 256, 0, stream>>>(xyz, points, out, x0);
  pad_w_kernel<<<(64 * 8 + 255) / 256, 256, 0, stream>>>(W1, wpad, 64, 6, 8);

  const int JT = NJ_ / 64 / 8;  // 1024 blocks in x (8 waves x 64-col tiles)

  // Layer 1: 8(pad) -> 64
  mlp_matmul_wmma<<<dim3(JT, 64 / 16), 256, 0, stream>>>(x0, wpad, b1, yA, 8);
  zero_kernel<<<1, 256, 0, stream>>>(stats, 256);
  channel_stats_kernel<<<dim3(64, 64), 256, 0, stream>>>(yA, sums, sumsq);
  bn_relu_kernel<<<dim3(512, 64), 256, 0, stream>>>(yA, sums, sumsq, g1, be1);

  // Layer 2: 64 -> 64
  mlp_matmul_wmma<<<dim3(JT, 64 / 16), 256, 0, stream>>>(yA, W2, b2, yB, 64);
  zero_kernel<<<1, 256, 0, stream>>>(stats, 256);
  channel_stats_kernel<<<dim3(64, 64), 256, 0, stream>>>(yB, sums, sumsq);
  bn_relu_kernel<<<dim3(512, 64), 256, 0, stream>>>(yB, sums, sumsq, g2, be2);

  // Layer 3: 64 -> 128, fused BN+ReLU+max epilogue to d_out
  mlp_matmul_wmma<<<dim3(JT, 128 / 16), 256, 0, stream>>>(yB, W3, b3, yC, 64);
  zero_kernel<<<1, 256, 0, stream>>>(stats, 256);
  channel_stats_kernel<<<dim3(128, 64), 256, 0, stream>>>(yC, sums, sumsq);
  bn_relu_max_kernel<<<dim3((B_ * M_) / 256, 128), 256, 0, stream>>>(
      yC, sums, sumsq, g3, be3, out);

  (void)in_sizes; (void)n_in; (void)out_size; (void)ws_size;
}